// local_emb_D_17205638988465
// MI455X (gfx1250) — compile-verified
//
#include <hip/hip_runtime.h>

typedef __attribute__((ext_vector_type(2))) float v2f;
typedef __attribute__((ext_vector_type(8))) float v8f;

#define H 128           // embedding dim (fixed by reference)
#define WAVE 32

// ---------------------------------------------------------------------------
// Phase 1: invn[n] = 1 / max(||emb[n,:]||_2, 1e-12)
// One wave per row: 32 lanes x float4 = 512B fully-coalesced read per row.
// ---------------------------------------------------------------------------
__global__ __launch_bounds__(256) void row_invnorm_kernel(
    const float* __restrict__ emb, float* __restrict__ invn, int N) {
  const int lane = threadIdx.x & (WAVE - 1);
  const int row  = (blockIdx.x * blockDim.x + threadIdx.x) >> 5;
  if (row >= N) return;

  const float4* r = reinterpret_cast<const float4*>(emb + (size_t)row * H);
  float4 v = r[lane];
  float s = v.x * v.x + v.y * v.y + v.z * v.z + v.w * v.w;
#pragma unroll
  for (int off = 16; off > 0; off >>= 1) s += __shfl_xor(s, off, WAVE);

  if (lane == 0) {
    float n = fmaxf(sqrtf(s), 1e-12f);
    invn[row] = 1.0f / n;
  }
}

// ---------------------------------------------------------------------------
// Phase 2: 16 edges per wave. Accumulate the 16x16 cross-Gram of
// (src rows scaled by d) x (dst rows) via 32x V_WMMA_F32_16X16X4_F32 over
// K=128, then read the diagonal = the 16 edge dot products.
//
// Fragment addressing (wave32, f32 16x16x4):
//   lane l, e = edge_base + (l & 15), col = (l >= 16) ? 2 : 0
//   A frag = float2 emb[src[e]]*d at k = 4t+col   (VGPR0=k, VGPR1=k+1)
//   B frag = float2 emb[dst[e]]   at k = 4t+col   (VGPR0=k, VGPR1=k+1)
// Per-edge scalars scale*invn[src]*invn[dst] factor out of the bilinear form
// and are applied to the diagonal at the end.
// ---------------------------------------------------------------------------
__global__ __launch_bounds__(256) void edge_dot_wmma_kernel(
    const float* __restrict__ emb, const int* __restrict__ src,
    const int* __restrict__ dst, const float* __restrict__ dvec,
    const float* __restrict__ scale, const float* __restrict__ invn,
    float* __restrict__ out, int E) {
  const int lane      = threadIdx.x & (WAVE - 1);
  const int waveInBlk = threadIdx.x >> 5;
  const long long waveId = (long long)blockIdx.x * 8 + waveInBlk;
  const long long ebase  = waveId * 16;
  if (ebase >= E) return;  // uniform across the wave

  const int myEdge = (int)ebase + (lane & 15);
  const int eClamp = (myEdge < E) ? myEdge : (E - 1);  // keep EXEC all-1s

  const int si = src[eClamp];
  const int di = dst[eClamp];
  const int col = (lane >> 4) << 1;  // 0 for lanes 0-15, 2 for lanes 16-31

  const float* arow = emb + (size_t)si * H + col;
  const float* brow = emb + (size_t)di * H + col;
  const float* drow = dvec + col;

  v8f c = {0.f, 0.f, 0.f, 0.f, 0.f, 0.f, 0.f, 0.f};
#pragma unroll 8
  for (int t = 0; t < H / 4; ++t) {
    v2f a  = *reinterpret_cast<const v2f*>(arow + 4 * t);
    v2f dd = *reinterpret_cast<const v2f*>(drow + 4 * t);
    v2f b  = *reinterpret_cast<const v2f*>(brow + 4 * t);
    a = a * dd;  // fold diagonal d[k] into A
    c = __builtin_amdgcn_wmma_f32_16x16x4_f32(
        /*neg_a=*/false, a, /*neg_b=*/false, b,
        /*c_mod=*/(short)0, c, /*reuse_a=*/false, /*reuse_b=*/false);
  }

  // Diagonal extraction: i<8 -> lane i, VGPR i ; i>=8 -> lane i+16, VGPR i-8.
  const bool writer = (lane < 8) || (lane >= 24);
  const int  idx    = (lane < 8) ? lane : (lane - 24);  // VGPR slot
  float diag = c[0];
#pragma unroll
  for (int r = 1; r < 8; ++r) diag = (idx == r) ? c[r] : diag;

  if (writer) {
    const int e = (int)ebase + ((lane < 8) ? lane : (lane - 16));
    if (e < E) {
      const float f = scale[0] * invn[si] * invn[di];
      out[e] = diag * f;
    }
  }
}

// ---------------------------------------------------------------------------
extern "C" void kernel_launch(void* const* d_in, const int* in_sizes, int n_in,
                              void* d_out, int out_size, void* d_ws,
                              size_t ws_size, hipStream_t stream) {
  const float* emb   = (const float*)d_in[0];
  const int*   src   = (const int*)d_in[1];
  const int*   dst   = (const int*)d_in[2];
  const float* dvec  = (const float*)d_in[3];
  const float* scale = (const float*)d_in[4];
  float*       out   = (float*)d_out;
  float*       invn  = (float*)d_ws;  // N floats of scratch

  const int N = in_sizes[0] / H;
  const int E = in_sizes[1];

  // Phase 1: 8 rows per 256-thread block (one wave per row).
  const int blocks1 = (N + 7) / 8;
  row_invnorm_kernel<<<blocks1, 256, 0, stream>>>(emb, invn, N);

  // Phase 2: 8 waves per block * 16 edges per wave = 128 edges/block.
  const int blocks2 = (E + 127) / 128;
  edge_dot_wmma_kernel<<<blocks2, 256, 0, stream>>>(emb, src, dst, dvec, scale,
                                                    invn, out, E);
}